// DetBenchTrain_16441134809698
// MI455X (gfx1250) — compile-verified
//
#include <hip/hip_runtime.h>
#include <stdint.h>

// ---------------------------------------------------------------------------
// Detection post-process: per-batch top-K=5000 over ~9.94M anchor*class scores
// (radix-select via 2-level 11-bit histograms) + box/class gather + bitonic
// sort for descending order. Memory-bound: ~2 streaming passes over 318MB.
// CDNA5 paths: double-buffered async global->LDS staging (ASYNCcnt split
// issue/wait with s_wait_asynccnt 1), reversed second pass for 192MB-L2 reuse.
// ---------------------------------------------------------------------------

#define USE_ASYNC_LDS 1

namespace det {

constexpr int C       = 90;
constexpr int K       = 5000;
constexpr int A       = 9;
constexpr int B       = 8;
constexpr int HB      = 2048;       // 11-bit radix buckets
constexpr int TIE_CAP = 32768;      // per-batch tie buffer capacity
constexpr int SORT_N  = 8192;       // pow2 >= K for bitonic sort
constexpr int BLK     = 256;
constexpr int ITER    = 8;
constexpr int CHUNK   = BLK * 4 * ITER;   // 8192 elements per block

// anchors-per-level offsets (A * S * S, cumulative)
// S = {96,48,24,12,6} -> {82944,20736,5184,1296,324}, N = 110484
constexpr unsigned N_TOT        = 110484u;
constexpr unsigned SCORES_TOT   = N_TOT * (unsigned)C;   // 9,943,560 per batch

// --- workspace layout (32-bit words) ---------------------------------------
constexpr size_t W_HIST1 = 0;                                  // B*HB
constexpr size_t W_SCAL  = W_HIST1 + (size_t)B * HB;           // B*32 scalars
constexpr size_t W_CVAL  = W_SCAL + (size_t)B * 32;            // B*K cand vals
constexpr size_t W_CIDX  = W_CVAL + (size_t)B * K;             // B*K cand idx
constexpr size_t W_TVAL  = W_CIDX + (size_t)B * K;             // B*TIE_CAP
constexpr size_t W_TIDX  = W_TVAL + (size_t)B * TIE_CAP;       // B*TIE_CAP
// scalar slots per batch: 0=thresh1 1=aboveTotal 2=candPos 3=tiePos
//                         4=thresh2 5=above2

__device__ __forceinline__ unsigned umin_(unsigned a, unsigned b) { return a < b ? a : b; }

// monotonic key: larger key <=> larger float
__device__ __forceinline__ unsigned key_of(float x) {
  unsigned u = __float_as_uint(x);
  return (u & 0x80000000u) ? ~u : (u | 0x80000000u);
}
__device__ __forceinline__ float val_of(unsigned k) {
  unsigned u = (k & 0x80000000u) ? (k & 0x7FFFFFFFu) : ~k;
  return __uint_as_float(u);
}

template <int N>
__device__ __forceinline__ void wait_asynccnt() {
#if __has_builtin(__builtin_amdgcn_s_wait_asynccnt)
  __builtin_amdgcn_s_wait_asynccnt(N);
#else
  if constexpr (N == 0)
    asm volatile("s_wait_asynccnt 0" ::: "memory");
  else
    asm volatile("s_wait_asynccnt 1" ::: "memory");
#endif
}

// one b128 async copy global->LDS per lane (tracked by ASYNCcnt)
__device__ __forceinline__ void async_copy16(const float* g, float* ldsDst) {
  unsigned ldsAddr = (unsigned)(uintptr_t)ldsDst;
  asm volatile("global_load_async_to_lds_b128 %0, %1, off"
               : : "v"(ldsAddr), "v"(g) : "memory");
}

// ---------------------------------------------------------------------------
// Pass 1: per-batch histogram of top-11-bit radix keys.
// Streams tiles via double-buffered async global->LDS: issue tile t+1, then
// s_wait_asynccnt 1 so tile t is complete while t+1 stays in flight.
// ---------------------------------------------------------------------------
__global__ void __launch_bounds__(BLK) hist_pass(const float* __restrict__ src,
                                                 unsigned perBatch,
                                                 unsigned* __restrict__ ws) {
  __shared__ unsigned h[HB];
  __shared__ float stage[2][BLK * 4];
  for (int i = threadIdx.x; i < HB; i += blockDim.x) h[i] = 0u;
  __syncthreads();

  const unsigned b = blockIdx.y;
  const float* base = src + (size_t)b * perBatch;
  const unsigned e0 = blockIdx.x * (unsigned)CHUNK + threadIdx.x * 4u;
  float* myStage0 = &stage[0][threadIdx.x * 4];
  float* myStage1 = &stage[1][threadIdx.x * 4];

#if USE_ASYNC_LDS
  if (e0 < perBatch) async_copy16(base + e0, myStage0);   // prologue: tile 0
#endif

  for (int it = 0; it < ITER; ++it) {
    const unsigned ec = e0 + (unsigned)it * (BLK * 4);
    const unsigned en = ec + (BLK * 4);
    const bool vn = (it + 1 < ITER) && (en < perBatch);
#if USE_ASYNC_LDS
    // issue tile it+1 into the other buffer before consuming tile it
    if (vn) async_copy16(base + en, (it & 1) ? myStage0 : myStage1);
#endif
    if (ec < perBatch) {   // perBatch is always a multiple of 4
      float4 v;
#if USE_ASYNC_LDS
      if (vn) wait_asynccnt<1>(); else wait_asynccnt<0>();
      const float* sp = (it & 1) ? myStage1 : myStage0;
      v = *reinterpret_cast<const float4*>(sp);
#else
      v = *reinterpret_cast<const float4*>(base + ec);
#endif
      atomicAdd(&h[key_of(v.x) >> 21], 1u);
      atomicAdd(&h[key_of(v.y) >> 21], 1u);
      atomicAdd(&h[key_of(v.z) >> 21], 1u);
      atomicAdd(&h[key_of(v.w) >> 21], 1u);
    }
  }
  __syncthreads();
  unsigned* gh = ws + W_HIST1 + (size_t)b * HB;
  for (int i = threadIdx.x; i < HB; i += blockDim.x) {
    unsigned c = h[i];
    if (c) atomicAdd(&gh[i], c);
  }
}

// ---------------------------------------------------------------------------
// Per-batch suffix scan: find bucket T with  count(>T) < K <= count(>=T).
// ---------------------------------------------------------------------------
__global__ void scan_thresh(unsigned* __restrict__ ws) {
  const unsigned b = blockIdx.x;
  const unsigned* hist = ws + W_HIST1 + (size_t)b * HB;
  unsigned* sc = ws + W_SCAL + (size_t)b * 32;
  if (threadIdx.x == 0) {
    unsigned acc = 0, t = 0, above = 0;
    bool found = false;
    for (int i = HB - 1; i >= 0 && !found; --i) {
      unsigned c = hist[i];
      if (acc + c >= (unsigned)K) { t = (unsigned)i; above = acc; found = true; }
      else acc += c;
    }
    if (!found) { t = 0; above = acc; }
    sc[0] = t;
    sc[1] = above;
  }
}

// ---------------------------------------------------------------------------
// Pass 2: re-stream; bucket>T -> candidate list; bucket==T -> tie buffer.
// Chunks are traversed in REVERSE order (and levels launched 4..0) so this
// pass begins on the data pass 1 touched last -- still resident in the
// 192MB L2 -- recovering ~60% of second-pass traffic as L2 hits.
// logical index = n*C + c with n = nOff + (h*S+w)*A + a  (matches the JAX
// transpose(0,2,3,1)+reshape concat ordering).
// ---------------------------------------------------------------------------
__global__ void __launch_bounds__(BLK) pass2(const float* __restrict__ src,
                                             unsigned perBatch, unsigned S,
                                             unsigned nOff,
                                             unsigned* __restrict__ ws) {
  const unsigned b = blockIdx.y;
  unsigned* sc = ws + W_SCAL + (size_t)b * 32;
  const unsigned T = sc[0];
  float*    cval = reinterpret_cast<float*>(ws + W_CVAL) + (size_t)b * K;
  unsigned* cidx = ws + W_CIDX + (size_t)b * K;
  float*    tval = reinterpret_cast<float*>(ws + W_TVAL) + (size_t)b * TIE_CAP;
  unsigned* tidx = ws + W_TIDX + (size_t)b * TIE_CAP;

  const float* base = src + (size_t)b * perBatch;
  const unsigned SS = S * S;
  const unsigned chunkId = gridDim.x - 1u - blockIdx.x;   // reversed traversal
  const unsigned start = chunkId * (unsigned)CHUNK;

  for (int it = 0; it < ITER; ++it) {
    unsigned e = start + (unsigned)it * (BLK * 4) + threadIdx.x * 4u;
    if (e < perBatch) {
      float4 v = *reinterpret_cast<const float4*>(base + e);
      float vv[4] = {v.x, v.y, v.z, v.w};
#pragma unroll
      for (int j = 0; j < 4; ++j) {
        unsigned key = key_of(vv[j]);
        unsigned bkt = key >> 21;
        if (bkt < T) continue;
        unsigned r   = e + (unsigned)j;            // (a*C+c, h, w) flattened
        unsigned ch  = r / SS;
        unsigned rem = r - ch * SS;
        unsigned hh  = rem / S;
        unsigned wp  = rem - hh * S;
        unsigned a   = ch / (unsigned)C;
        unsigned c   = ch - a * (unsigned)C;
        unsigned n   = nOff + (hh * S + wp) * (unsigned)A + a;
        unsigned logical = n * (unsigned)C + c;
        if (bkt > T) {
          unsigned p = atomicAdd(&sc[2], 1u);
          if (p < (unsigned)K) { cval[p] = vv[j]; cidx[p] = logical; }
        } else {
          unsigned p = atomicAdd(&sc[3], 1u);
          if (p < (unsigned)TIE_CAP) { tval[p] = vv[j]; tidx[p] = logical; }
        }
      }
    }
  }
}

// ---------------------------------------------------------------------------
// Refine within the threshold bucket using the next 11 key bits.
// ---------------------------------------------------------------------------
__global__ void __launch_bounds__(BLK) refine_ties(unsigned* __restrict__ ws) {
  __shared__ unsigned h[HB];
  const unsigned b = blockIdx.x;
  unsigned* sc = ws + W_SCAL + (size_t)b * 32;
  for (int i = threadIdx.x; i < HB; i += blockDim.x) h[i] = 0u;
  __syncthreads();

  const float* tval = reinterpret_cast<const float*>(ws + W_TVAL) + (size_t)b * TIE_CAP;
  const unsigned M = umin_(sc[3], (unsigned)TIE_CAP);
  for (unsigned i = threadIdx.x; i < M; i += blockDim.x) {
    unsigned key = key_of(tval[i]);
    atomicAdd(&h[(key >> 10) & (HB - 1)], 1u);
  }
  __syncthreads();

  if (threadIdx.x == 0) {
    unsigned above = umin_(sc[1], (unsigned)K);
    unsigned need  = (unsigned)K - above;
    unsigned acc = 0, t2 = 0, above2 = 0;
    bool found = false;
    for (int i = HB - 1; i >= 0 && !found; --i) {
      unsigned c = h[i];
      if (acc + c >= need) { t2 = (unsigned)i; above2 = acc; found = true; }
      else acc += c;
    }
    if (!found) { t2 = 0; above2 = acc; }
    sc[4] = t2;
    sc[5] = above2;
  }
}

// ---------------------------------------------------------------------------
// Fill candidate slots [above .. K) from the tie buffer; pad leftovers.
// One block per batch so LDS counters give deterministic-capacity fill.
// ---------------------------------------------------------------------------
__global__ void __launch_bounds__(BLK) select_ties(unsigned* __restrict__ ws) {
  __shared__ unsigned cntA, cntE;
  const unsigned b = blockIdx.x;
  unsigned* sc = ws + W_SCAL + (size_t)b * 32;
  if (threadIdx.x == 0) { cntA = 0u; cntE = 0u; }
  __syncthreads();

  float*    cval = reinterpret_cast<float*>(ws + W_CVAL) + (size_t)b * K;
  unsigned* cidx = ws + W_CIDX + (size_t)b * K;
  const float*    tval = reinterpret_cast<const float*>(ws + W_TVAL) + (size_t)b * TIE_CAP;
  const unsigned* tidx = ws + W_TIDX + (size_t)b * TIE_CAP;

  const unsigned above = umin_(sc[1], (unsigned)K);
  const unsigned M     = umin_(sc[3], (unsigned)TIE_CAP);
  const unsigned t2    = sc[4];

  // phase 1: ties strictly above second-level threshold
  for (unsigned i = threadIdx.x; i < M; i += blockDim.x) {
    unsigned b2 = (key_of(tval[i]) >> 10) & (HB - 1);
    if (b2 > t2) {
      unsigned p = atomicAdd(&cntA, 1u);
      unsigned slot = above + p;
      if (slot < (unsigned)K) { cval[slot] = tval[i]; cidx[slot] = tidx[i]; }
    }
  }
  __syncthreads();
  const unsigned base2 = above + cntA;

  // phase 2: ties equal to second-level threshold (arbitrary tie-break)
  for (unsigned i = threadIdx.x; i < M; i += blockDim.x) {
    unsigned b2 = (key_of(tval[i]) >> 10) & (HB - 1);
    if (b2 == t2) {
      unsigned p = atomicAdd(&cntE, 1u);
      unsigned slot = base2 + p;
      if (slot < (unsigned)K) { cval[slot] = tval[i]; cidx[slot] = tidx[i]; }
    }
  }
  __syncthreads();

  unsigned filled = base2 + cntE;
  if (filled > (unsigned)K) filled = (unsigned)K;
  for (unsigned s = filled + threadIdx.x; s < (unsigned)K; s += blockDim.x) {
    cval[s] = -3.0e38f;
    cidx[s] = 0xFFFFFFFFu;
  }
}

// ---------------------------------------------------------------------------
// Per-batch bitonic sort (descending by value, ascending by index on ties)
// of the K candidates in LDS, then write all four outputs with box gather.
// ---------------------------------------------------------------------------
__global__ void __launch_bounds__(512) sort_and_output(
    const unsigned* __restrict__ ws, float* __restrict__ out,
    const float* __restrict__ box0, const float* __restrict__ box1,
    const float* __restrict__ box2, const float* __restrict__ box3,
    const float* __restrict__ box4) {
  __shared__ unsigned long long sk[SORT_N];   // 64 KB
  const unsigned b = blockIdx.x;
  const float*    cval = reinterpret_cast<const float*>(ws + W_CVAL) + (size_t)b * K;
  const unsigned* cidx = ws + W_CIDX + (size_t)b * K;

  for (unsigned i = threadIdx.x; i < (unsigned)SORT_N; i += blockDim.x) {
    unsigned long long comp = 0ull;
    if (i < (unsigned)K) {
      unsigned key = key_of(cval[i]);
      unsigned idx = cidx[i];
      comp = ((unsigned long long)key << 32) | (unsigned)(~idx);  // desc val, asc idx
    }
    sk[i] = comp;
  }
  __syncthreads();

  for (unsigned kk = 2; kk <= (unsigned)SORT_N; kk <<= 1) {
    for (unsigned j = kk >> 1; j > 0; j >>= 1) {
      for (unsigned i = threadIdx.x; i < (unsigned)SORT_N; i += blockDim.x) {
        unsigned ixj = i ^ j;
        if (ixj > i) {
          unsigned long long x = sk[i], y = sk[ixj];
          bool descBlock = ((i & kk) == 0u);
          bool sw = descBlock ? (x < y) : (x > y);
          if (sw) { sk[i] = y; sk[ixj] = x; }
        }
      }
      __syncthreads();
    }
  }

  const unsigned offs[6] = {0u, 82944u, 103680u, 108864u, 110160u, 110484u};
  const unsigned feats[5] = {96u, 48u, 24u, 12u, 6u};
  const float* boxes[5] = {box0, box1, box2, box3, box4};

  for (unsigned r = threadIdx.x; r < (unsigned)K; r += blockDim.x) {
    unsigned long long comp = sk[r];
    unsigned key = (unsigned)(comp >> 32);
    unsigned idx = ~((unsigned)comp);
    float val = val_of(key);
    unsigned n = 0, c = 0;
    float bx[4] = {0.f, 0.f, 0.f, 0.f};
    if (idx < SCORES_TOT) {
      n = idx / (unsigned)C;
      c = idx % (unsigned)C;
      int l = 0;
      while (l < 4 && n >= offs[l + 1]) ++l;
      unsigned loc = n - offs[l];
      unsigned S = feats[l];
      unsigned hw = loc / (unsigned)A;
      unsigned a  = loc % (unsigned)A;
      unsigned hh = hw / S, wp = hw % S;
      const float* bp = boxes[l] + (size_t)b * (36u * S * S);
#pragma unroll
      for (int j = 0; j < 4; ++j)
        bx[j] = bp[(((a * 4u + (unsigned)j) * S) + hh) * S + wp];
    } else {
      val = 0.0f;   // defensive: should not happen
    }
    // outputs concatenated flat: cls_topk[B,K,1], box_topk[B,K,4],
    // indices[B,K], classes[B,K] (ints written as exact floats)
    out[(size_t)b * K + r] = val;
    float* obox = out + (size_t)B * K;
#pragma unroll
    for (int j = 0; j < 4; ++j)
      obox[((size_t)b * K + r) * 4 + j] = bx[j];
    out[(size_t)B * K * 5 + (size_t)b * K + r] = (float)n;
    out[(size_t)B * K * 6 + (size_t)b * K + r] = (float)c;
  }
}

}  // namespace det

extern "C" void kernel_launch(void* const* d_in, const int* in_sizes, int n_in,
                              void* d_out, int out_size, void* d_ws, size_t ws_size,
                              hipStream_t stream) {
  (void)in_sizes; (void)n_in; (void)out_size; (void)ws_size;
  // setup_inputs() dict order is interleaved: cls_out_l, box_out_l per level.
  const float* cls[5];
  const float* box[5];
  for (int l = 0; l < 5; ++l) {
    cls[l] = (const float*)d_in[2 * l + 0];
    box[l] = (const float*)d_in[2 * l + 1];
  }
  unsigned* ws = (unsigned*)d_ws;
  float* out = (float*)d_out;

  static const unsigned S[5]    = {96u, 48u, 24u, 12u, 6u};
  static const unsigned nOff[5] = {0u, 82944u, 103680u, 108864u, 110160u};

  // zero histograms + per-batch scalars
  hipMemsetAsync(ws, 0, det::W_CVAL * sizeof(unsigned), stream);

  // pass 1: levels 0..4 (forward stream)
  for (int l = 0; l < 5; ++l) {
    unsigned per = 810u * S[l] * S[l];   // A*C * S * S
    dim3 grid((per + det::CHUNK - 1) / det::CHUNK, det::B);
    det::hist_pass<<<grid, dim3(det::BLK), 0, stream>>>(cls[l], per, ws);
  }

  det::scan_thresh<<<dim3(det::B), dim3(64), 0, stream>>>(ws);

  // pass 2: levels 4..0 with reversed chunks -> re-read L2-resident tail first
  for (int l = 4; l >= 0; --l) {
    unsigned per = 810u * S[l] * S[l];
    dim3 grid((per + det::CHUNK - 1) / det::CHUNK, det::B);
    det::pass2<<<grid, dim3(det::BLK), 0, stream>>>(cls[l], per, S[l], nOff[l], ws);
  }

  det::refine_ties<<<dim3(det::B), dim3(det::BLK), 0, stream>>>(ws);
  det::select_ties<<<dim3(det::B), dim3(det::BLK), 0, stream>>>(ws);
  det::sort_and_output<<<dim3(det::B), dim3(512), 0, stream>>>(
      ws, out, box[0], box[1], box[2], box[3], box[4]);
}